// P_Model_49855980372254
// MI455X (gfx1250) — compile-verified
//
#include <hip/hip_runtime.h>

typedef float v2f __attribute__((ext_vector_type(2)));
typedef float v8f __attribute__((ext_vector_type(8)));

#define NN 4096      // N
#define NB 512       // batch
#define DT_ 0.1f

#define CH   32                 // K-chunk rows staged per iteration
#define LSTR 160                // LDS pair-row stride in floats (160 % 64 == 32)
#define ASTR ((size_t)(2 * NN)) // P22I pair-row stride in floats

__device__ __forceinline__ float dot4f(float4 a, float4 b) {
    return a.x*b.x + a.y*b.y + a.z*b.z + a.w*b.w;
}

// ---------------------------------------------------------------------------
// Build P22 transposed + K-pair-interleaved:
//   P22I[(k>>1)*2N + 2m + (k&1)] = P22[m][k]
//                                = cb*M[m]*Z[k] + DT*(U[m]·V[k] + W[m][k]) + (1-DT)*(k==m)
// Pair interleave makes the WMMA A-fragment a single b64 per lane.
// W is read coalesced and transposed through LDS.
// ---------------------------------------------------------------------------
__global__ void __launch_bounds__(256) build_p22i_kernel(
    const float* __restrict__ W, const float* __restrict__ U,
    const float* __restrict__ V, const float* __restrict__ Mv,
    const float* __restrict__ Zv, const float* __restrict__ Cm,
    const float* __restrict__ Bm, float* __restrict__ P22I)
{
    __shared__ float tile[16][65];           // tile[k_local][m_local] = W[m][k]
    const int t  = threadIdx.x;
    const int k0 = blockIdx.x * 16;
    const int m0 = blockIdx.y * 64;

    {   // coalesced read of W[m0..m0+63][k0..k0+15], transpose into LDS
        const int ml = t >> 2;               // 0..63
        const int kq = (t & 3) << 2;         // 0,4,8,12
        float4 w4 = *(const float4*)&W[(size_t)(m0 + ml) * NN + k0 + kq];
        tile[kq + 0][ml] = w4.x;
        tile[kq + 1][ml] = w4.y;
        tile[kq + 2][ml] = w4.z;
        tile[kq + 3][ml] = w4.w;
    }
    __syncthreads();

    const float cb = Cm[0] * Bm[0] + Cm[1] * Bm[1];   // C @ Bmat (scalar)
    const int p  = t >> 5;                   // pair row 0..7
    const int c2 = (t & 31) << 1;            // m_local 0..62 (even)
    const int k  = k0 + 2 * p;
    const int m  = m0 + c2;

    const float  zk0 = Zv[k],   zk1 = Zv[k + 1];
    const float4 vk0 = *(const float4*)&V[k * 4];
    const float4 vk1 = *(const float4*)&V[(k + 1) * 4];

    const float4 um0 = *(const float4*)&U[m * 4];
    const float4 um1 = *(const float4*)&U[(m + 1) * 4];
    const float  mm0 = Mv[m], mm1 = Mv[m + 1];

    float v00 = cb * mm0 * zk0 + DT_ * (dot4f(um0, vk0) + tile[2*p    ][c2    ]);
    float v10 = cb * mm0 * zk1 + DT_ * (dot4f(um0, vk1) + tile[2*p + 1][c2    ]);
    float v01 = cb * mm1 * zk0 + DT_ * (dot4f(um1, vk0) + tile[2*p    ][c2 + 1]);
    float v11 = cb * mm1 * zk1 + DT_ * (dot4f(um1, vk1) + tile[2*p + 1][c2 + 1]);
    if (k     == m    ) v00 += (1.0f - DT_);
    if (k + 1 == m    ) v10 += (1.0f - DT_);
    if (k     == m + 1) v01 += (1.0f - DT_);
    if (k + 1 == m + 1) v11 += (1.0f - DT_);

    float4 outv = make_float4(v00, v10, v01, v11);   // {.. 2m+0, 2m+1, 2m+2, 2m+3}
    *(float4*)&P22I[(size_t)((k0 >> 1) + p) * ASTR + 2 * m] = outv;
}

// ---------------------------------------------------------------------------
// Per-step small work: zh = Z·h, con[t] = Bmat[1]*zh, and (if not last step)
// the two x-rows of the next state.
// ---------------------------------------------------------------------------
__global__ void __launch_bounds__(256) small_step_kernel(
    const float* __restrict__ traj_t, float* __restrict__ con_t,
    float* __restrict__ traj_next,
    const float* __restrict__ Zv, const float* __restrict__ Amat,
    const float* __restrict__ Bm)
{
    const int n = blockIdx.x * blockDim.x + threadIdx.x;   // 0..511
    const float* __restrict__ h = traj_t + 2 * NB;
    float zh = 0.0f;
#pragma unroll 8
    for (int k = 0; k < NN; ++k)
        zh = fmaf(Zv[k], h[(size_t)k * NB + n], zh);
    con_t[n] = Bm[1] * zh;
    if (traj_next != nullptr) {
        const float x0 = traj_t[n];
        const float x1 = traj_t[NB + n];
        traj_next[n]      = Amat[0]*x0 + Amat[1]*x1 + Bm[0]*zh;
        traj_next[NB + n] = Amat[2]*x0 + Amat[3]*x1 + Bm[1]*zh;
    }
}

// ---------------------------------------------------------------------------
// h_next = P22 @ h + P21 @ x  via V_WMMA_F32_16X16X4_F32.
// 8 waves/block; wave w owns M-tile [m0,m0+16), block shares N-range [nb,nb+64).
// Fragment layout (ISA 16x4 f32): VGPR v, lane l -> K = kbase + v + 2*(l>=16),
// M/N = l&15.  Pair interleaving makes A one global b64 and B one ds b64.
// C/D layout: VGPR v, lane l -> M = v + 8*(l>=16), N = l&15.
// ---------------------------------------------------------------------------
__global__ void __launch_bounds__(256) h_step_wmma_kernel(
    const float* __restrict__ P22I, const float* __restrict__ traj_t,
    float* __restrict__ traj_next,
    const float* __restrict__ Mv, const float* __restrict__ Amat,
    const float* __restrict__ Cm)
{
    __shared__ float Bt[(CH / 2) * LSTR];         // 16 pair-rows * 160 = 10 KB
    const int tid  = threadIdx.x;
    const int lane = tid & 31;
    const int wave = tid >> 5;
    const int l15  = lane & 15;
    const int phal = lane >> 4;                   // 0 or 1 (lane half -> +1 pair row)
    const int m0   = blockIdx.x * 128 + wave * 16;
    const int nb   = blockIdx.y * 64;
    const float* __restrict__ h = traj_t + 2 * NB;

    v8f acc0 = {}, acc1 = {}, acc2 = {}, acc3 = {};

    // ---- staging mapping: thread handles pair-rows sp and sp+8, cols sc,sc+1
    const int sp = tid >> 5;                      // 0..7
    const int sc = (tid & 31) << 1;               // 0..62
    const float* hp0 = h + (size_t)(2 * sp) * NB + nb + sc;      // row 2sp
    const float* hp1 = hp0 + NB;                                 // row 2sp+1
    const float* hp2 = hp0 + (size_t)16 * NB;                    // row 16+2sp
    const float* hp3 = hp2 + NB;

    // ---- A-fragment lane base (pair-interleaved P22I)
    const int acol = m0 + l15;
    const float* aLane = P22I + (size_t)phal * ASTR + 2 * acol;

    // preload chunk 0
    v2f ra0 = *(const v2f*)hp0;  v2f rb0 = *(const v2f*)hp1;
    v2f ra1 = *(const v2f*)hp2;  v2f rb1 = *(const v2f*)hp3;

    const int nch = NN / CH;                      // 128 chunks
    for (int c = 0; c < nch; ++c) {
        __syncthreads();
        *(float4*)&Bt[sp * LSTR + 2 * sc]       = make_float4(ra0.x, rb0.x, ra0.y, rb0.y);
        *(float4*)&Bt[(sp + 8) * LSTR + 2 * sc] = make_float4(ra1.x, rb1.x, ra1.y, rb1.y);
        if (c + 1 < nch) {
            const size_t adv = (size_t)(c + 1) * CH * NB;
            ra0 = *(const v2f*)(hp0 + adv);  rb0 = *(const v2f*)(hp1 + adv);
            ra1 = *(const v2f*)(hp2 + adv);  rb1 = *(const v2f*)(hp3 + adv);
            // pull next chunk's A stream toward the WGP (global_prefetch_b8)
            __builtin_prefetch(aLane + (size_t)((c + 1) * 16) * ASTR, 0, 1);
        }
        __syncthreads();
#pragma unroll
        for (int kk = 0; kk < CH; kk += 4) {
            const v2f a = *(const v2f*)(aLane + (size_t)(c * 16 + (kk >> 1)) * ASTR);
            const int bbase = ((kk >> 1) + phal) * LSTR + (l15 << 1);
            const v2f b0 = *(const v2f*)&Bt[bbase +  0];
            const v2f b1 = *(const v2f*)&Bt[bbase + 32];
            const v2f b2 = *(const v2f*)&Bt[bbase + 64];
            const v2f b3 = *(const v2f*)&Bt[bbase + 96];
            acc0 = __builtin_amdgcn_wmma_f32_16x16x4_f32(false, a, false, b0, (short)0, acc0, false, false);
            acc1 = __builtin_amdgcn_wmma_f32_16x16x4_f32(false, a, false, b1, (short)0, acc1, false, false);
            acc2 = __builtin_amdgcn_wmma_f32_16x16x4_f32(false, a, false, b2, (short)0, acc2, false, false);
            acc3 = __builtin_amdgcn_wmma_f32_16x16x4_f32(false, a, false, b3, (short)0, acc3, false, false);
        }
    }

    // Epilogue: += P21 @ x  with P21[m][j] = DT*M[m]*ca[j], ca = C @ A.
    const float ca0 = Cm[0]*Amat[0] + Cm[1]*Amat[2];
    const float ca1 = Cm[0]*Amat[1] + Cm[1]*Amat[3];
    const int mh = phal << 3;                     // 0 or 8
    float xc[4];
#pragma unroll
    for (int ti = 0; ti < 4; ++ti) {
        const int n = nb + ti * 16 + l15;
        xc[ti] = ca0 * traj_t[n] + ca1 * traj_t[NB + n];
    }
#pragma unroll
    for (int v = 0; v < 8; ++v) {
        const int m = m0 + mh + v;
        const float pm = DT_ * Mv[m];
        float* __restrict__ orow = traj_next + (size_t)(2 + m) * NB + nb + l15;
        orow[0]  = acc0[v] + pm * xc[0];
        orow[16] = acc1[v] + pm * xc[1];
        orow[32] = acc2[v] + pm * xc[2];
        orow[48] = acc3[v] + pm * xc[3];
    }
}

// ---------------------------------------------------------------------------
extern "C" void kernel_launch(void* const* d_in, const int* in_sizes, int n_in,
                              void* d_out, int out_size, void* d_ws, size_t ws_size,
                              hipStream_t stream)
{
    const float* x0 = (const float*)d_in[0];
    const float* x1 = (const float*)d_in[1];
    const float* h0 = (const float*)d_in[2];
    const float* A  = (const float*)d_in[3];
    const float* Bm = (const float*)d_in[4];
    const float* Cm = (const float*)d_in[5];
    const float* Mv = (const float*)d_in[6];
    const float* Zv = (const float*)d_in[7];
    const float* U  = (const float*)d_in[8];
    const float* V  = (const float*)d_in[9];
    const float* W  = (const float*)d_in[10];
    (void)in_sizes; (void)n_in; (void)ws_size;

    const int stateN = (NN + 2) * NB;                       // 4098*512
    const int steps  = out_size / (stateN + NB);            // 64
    float* traj = (float*)d_out;                            // (steps, 4098, 512)
    float* con  = traj + (size_t)steps * stateN;            // (steps, 512)
    float* P22I = (float*)d_ws;                             // 4096*4096 f32 (pair-interleaved)

    // traj[0] = state0 = [x0; x1; h0]
    hipMemcpyAsync(traj,          x0, NB * sizeof(float),            hipMemcpyDeviceToDevice, stream);
    hipMemcpyAsync(traj + NB,     x1, NB * sizeof(float),            hipMemcpyDeviceToDevice, stream);
    hipMemcpyAsync(traj + 2*NB,   h0, (size_t)NN*NB*sizeof(float),   hipMemcpyDeviceToDevice, stream);

    build_p22i_kernel<<<dim3(NN/16, NN/64), 256, 0, stream>>>(W, U, V, Mv, Zv, Cm, Bm, P22I);

    for (int t = 0; t < steps; ++t) {
        const float* cur = traj + (size_t)t * stateN;
        float* nxt = (t + 1 < steps) ? traj + (size_t)(t + 1) * stateN : nullptr;
        small_step_kernel<<<dim3(NB/256), 256, 0, stream>>>(cur, con + (size_t)t*NB, nxt, Zv, A, Bm);
        if (nxt != nullptr)
            h_step_wmma_kernel<<<dim3(NN/128, NB/64), 256, 0, stream>>>(P22I, cur, nxt, Mv, A, Cm);
    }
}